// FeatureLoss_48524540510511
// MI455X (gfx1250) — compile-verified
//
#include <hip/hip_runtime.h>
#include <math.h>

// ---------------- problem constants ----------------
#define NB   8
#define CC   256
#define CH   128           // CC/2
#define HH   100
#define WW   152
#define HWP  (HH*WW)       // 15200
#define MB   20
#define TILES ((HWP+255)/256)  // 60
#define NHW  (NB*HWP)
#define TEMP_INV 2.0f      // 1/0.5
#define K_ALPHA 0.001f
#define K_BETA  0.0005f
#define K_GAMMA 0.001f
#define K_LAMB  5e-06f
#define LNEPS   1e-5f

// ---------------- workspace layout (floats) ----------------
#define OFF_A0   0               // feaT  -> S_t   [N,HW]
#define OFF_A1   (NHW)           // feaS  -> S_s   [N,HW]
#define OFF_A2   (2*NHW)         // mS    -> a_s   [N,HW]
#define OFF_A3   (3*NHW)         // mT    -> a_t   [N,HW]
#define OFF_MFG  (4*NHW)         // Mask_fg        [N,HW]
#define OFF_CHT  (5*NHW)         // chT -> C_t     [N,C]
#define OFF_CHS  (OFF_CHT + NB*CC)
#define OFF_DSUM (OFF_CHS + NB*CC)   // sum_hw (S-T)   [N,C]
#define OFF_CTXS (OFF_DSUM + NB*CC)
#define OFF_CTXT (OFF_CTXS + NB*CC)
#define OFF_BG   (OFF_CTXT + NB*CC)  // bg pixel count [N]
#define OFF_SCAL (OFF_BG + NB)       // 0:dsq 1:fg 2:bg 3:mask

// ---------------- reductions ----------------
__device__ __forceinline__ float wave_sum(float v){
#pragma unroll
  for(int o=16;o>0;o>>=1) v += __shfl_xor(v,o,32);
  return v;
}
__device__ __forceinline__ float wave_max(float v){
#pragma unroll
  for(int o=16;o>0;o>>=1) v = fmaxf(v, __shfl_xor(v,o,32));
  return v;
}
// block-wide sum, broadcast to all threads; red needs NT/32+1 slots
template<int NT>
__device__ __forceinline__ float block_sum(float v, float* red){
  const int lane = threadIdx.x & 31, w = threadIdx.x >> 5;
  const int NW = NT/32;
  v = wave_sum(v);
  __syncthreads();
  if(lane==0) red[w] = v;
  __syncthreads();
  float t = ((int)threadIdx.x < NW) ? red[threadIdx.x] : 0.f;
  if(w==0){ t = wave_sum(t); if(lane==0) red[NW] = t; }
  __syncthreads();
  return red[NW];
}
template<int NT>
__device__ __forceinline__ float block_max(float v, float* red){
  const int lane = threadIdx.x & 31, w = threadIdx.x >> 5;
  const int NW = NT/32;
  v = wave_max(v);
  __syncthreads();
  if(lane==0) red[w] = v;
  __syncthreads();
  float t = ((int)threadIdx.x < NW) ? red[threadIdx.x] : -INFINITY;
  if(w==0){ t = wave_max(t); if(lane==0) red[NW] = t; }
  __syncthreads();
  return red[NW];
}

// ---------------- kernels ----------------
__global__ void k_init(float* ws){
  int i = threadIdx.x;
  if(i < 8)  ws[OFF_SCAL+i] = 0.f;
  if(i < NB) ws[OFF_BG+i]   = 0.f;
}

// pass 1, hw-major: per-pixel channel means (|T|,|S|), spatial-pool logits, global sum d^2
__global__ void k_pass1_hw(const float* __restrict__ pS, const float* __restrict__ pT,
                           const float* __restrict__ wms, const float* __restrict__ bms,
                           const float* __restrict__ wmt, const float* __restrict__ bmt,
                           float* __restrict__ ws){
  __shared__ float lms[CC], lmt[CC];
  __shared__ float red[16];
  const int n = blockIdx.x / TILES, tile = blockIdx.x % TILES;
  const int tid = threadIdx.x;
  lms[tid] = wms[tid]; lmt[tid] = wmt[tid];
  __syncthreads();
  const int hw = tile*256 + tid;
  const bool ok = hw < HWP;
  const size_t base = (size_t)n*CC*HWP + hw;
  float aT=0.f, aS=0.f, aMS=0.f, aMT=0.f, d2=0.f;
  for(int c=0;c<CC;++c){
    float sv = ok ? pS[base + (size_t)c*HWP] : 0.f;
    float tv = ok ? pT[base + (size_t)c*HWP] : 0.f;
    aT += fabsf(tv); aS += fabsf(sv);
    aMS += sv*lms[c]; aMT += tv*lmt[c];
    float d = sv - tv; d2 += d*d;
  }
  if(ok){
    const int o = n*HWP + hw;
    ws[OFF_A0+o] = aT * (1.0f/CC);
    ws[OFF_A1+o] = aS * (1.0f/CC);
    ws[OFF_A2+o] = aMS + bms[0];
    ws[OFF_A3+o] = aMT + bmt[0];
  }
  float tot = block_sum<256>(d2, red);
  if(tid==0) atomicAdd(&ws[OFF_SCAL+0], tot);
}

// pass 1, c-major: per-channel spatial means and sum(S-T)
__global__ void k_pass1_nc(const float* __restrict__ pS, const float* __restrict__ pT,
                           float* __restrict__ ws){
  __shared__ float red[16];
  const int n = blockIdx.x / CC, c = blockIdx.x % CC;
  const size_t base = ((size_t)n*CC + c)*HWP;
  float sT=0.f, sS=0.f, sD=0.f;
  for(int hw=threadIdx.x; hw<HWP; hw+=256){
    float sv = pS[base+hw], tv = pT[base+hw];
    sT += fabsf(tv); sS += fabsf(sv); sD += sv - tv;
  }
  sT = block_sum<256>(sT, red);
  sS = block_sum<256>(sS, red);
  sD = block_sum<256>(sD, red);
  if(threadIdx.x==0){
    ws[OFF_CHT  + n*CC + c] = sT * (1.0f/HWP);
    ws[OFF_CHS  + n*CC + c] = sS * (1.0f/HWP);
    ws[OFF_DSUM + n*CC + c] = sD;
  }
}

// in-place softmax over HW for the 4 [N,HW] arrays (S uses temp + HW scale)
__global__ void k_softmax_hw(float* __restrict__ ws){
  __shared__ float red[16];
  const int arr = blockIdx.x & 3, n = blockIdx.x >> 2;
  float* buf = ws + arr*NHW + n*HWP;
  const float scale = (arr < 2) ? (float)HWP : 1.0f;
  const float it    = (arr < 2) ? TEMP_INV  : 1.0f;
  float m = -INFINITY;
  for(int i=threadIdx.x;i<HWP;i+=256) m = fmaxf(m, buf[i]);
  m = block_max<256>(m, red);
  float s = 0.f;
  for(int i=threadIdx.x;i<HWP;i+=256) s += expf((buf[i]-m)*it);
  s = block_sum<256>(s, red);
  const float inv = scale / s;
  for(int i=threadIdx.x;i<HWP;i+=256) buf[i] = expf((buf[i]-m)*it) * inv;
}

// channel softmax (C_t, C_s) + channel part of mask loss
__global__ void k_softmax_c(float* __restrict__ ws){
  __shared__ float red[16];
  const int n = blockIdx.x, c = threadIdx.x;
  float vT = ws[OFF_CHT + n*CC + c];
  float vS = ws[OFF_CHS + n*CC + c];
  float mT = block_max<256>(vT, red);
  float eT = expf((vT-mT)*TEMP_INV);
  float sT = block_sum<256>(eT, red);
  float Ct = (float)CC * eT / sT;
  float mS = block_max<256>(vS, red);
  float eS = expf((vS-mS)*TEMP_INV);
  float sS = block_sum<256>(eS, red);
  float Cs = (float)CC * eS / sS;
  ws[OFF_CHT + n*CC + c] = Ct;
  ws[OFF_CHS + n*CC + c] = Cs;
  float part = block_sum<256>(fabsf(Cs - Ct), red);
  if(c==0) atomicAdd(&ws[OFF_SCAL+3], part);
}

// fg mask from boxes, bg pixel count, spatial part of mask loss
__global__ void k_masks(const float* __restrict__ boxes, float* __restrict__ ws){
  __shared__ float red[16];
  __shared__ int   bhmin[MB], bhmax[MB], bwmin[MB], bwmax[MB];
  __shared__ float barea[MB];
  const int n = blockIdx.x / TILES, tile = blockIdx.x % TILES;
  const int tid = threadIdx.x;
  if(tid < MB){
    const float* b = boxes + (n*MB + tid)*4;
    float cx=b[0], cy=b[1], bw=b[2], bh=b[3];
    float x1=(cx-bw*0.5f)*WW, y1=(cy-bh*0.5f)*HH;
    float x2=(cx+bw*0.5f)*WW, y2=(cy+bh*0.5f)*HH;
    int wmn = (int)fminf(fmaxf(floorf(x1),0.f),(float)(WW-1));
    int wmx = (int)fminf(fmaxf(ceilf (x2),0.f),(float)(WW-1));
    int hmn = (int)fminf(fmaxf(floorf(y1),0.f),(float)(HH-1));
    int hmx = (int)fminf(fmaxf(ceilf (y2),0.f),(float)(HH-1));
    bhmin[tid]=hmn; bhmax[tid]=hmx; bwmin[tid]=wmn; bwmax[tid]=wmx;
    barea[tid] = 1.0f/((float)(hmx+1-hmn)*(float)(wmx+1-wmn));
  }
  __syncthreads();
  const int hw = tile*256 + tid;
  const bool ok = hw < HWP;
  float mfg=0.f, sdiff=0.f, bgc=0.f;
  if(ok){
    const int h = hw / WW, w = hw % WW;
    for(int m=0;m<MB;++m){
      if(h>=bhmin[m] && h<=bhmax[m] && w>=bwmin[m] && w<=bwmax[m])
        mfg = fmaxf(mfg, barea[m]);
    }
    ws[OFF_MFG + n*HWP + hw] = mfg;
    bgc   = (mfg==0.f) ? 1.f : 0.f;
    sdiff = fabsf(ws[OFF_A1+n*HWP+hw] - ws[OFF_A0+n*HWP+hw]);
  }
  float t = block_sum<256>(bgc, red);
  if(tid==0) atomicAdd(&ws[OFF_BG+n], t);
  t = block_sum<256>(sdiff, red);
  if(tid==0) atomicAdd(&ws[OFF_SCAL+3], t);
}

// pass 2, hw-major: fg/bg weighted squared-diff sums
__global__ void k_fgbg(const float* __restrict__ pS, const float* __restrict__ pT,
                       float* __restrict__ ws){
  __shared__ float ct[CC];
  __shared__ float red[16];
  const int n = blockIdx.x / TILES, tile = blockIdx.x % TILES;
  const int tid = threadIdx.x;
  ct[tid] = ws[OFF_CHT + n*CC + tid];     // C_t
  __syncthreads();
  const int hw = tile*256 + tid;
  const bool ok = hw < HWP;
  const size_t base = (size_t)n*CC*HWP + hw;
  float s2c = 0.f;
  for(int c=0;c<CC;++c){
    float sv = ok ? pS[base + (size_t)c*HWP] : 0.f;
    float tv = ok ? pT[base + (size_t)c*HWP] : 0.f;
    float d = sv - tv;
    s2c += d*d*ct[c];
  }
  float fg=0.f, bg=0.f;
  if(ok){
    float st  = ws[OFF_A0  + n*HWP + hw];   // S_t
    float mfg = ws[OFF_MFG + n*HWP + hw];
    float bgs = ws[OFF_BG  + n];
    float mbg = (mfg==0.f) ? ((bgs>0.f) ? 1.0f/bgs : 1.0f) : 0.f;
    fg = s2c * st * mfg;
    bg = s2c * st * mbg;
  }
  float t = block_sum<256>(fg, red);
  if(tid==0) atomicAdd(&ws[OFF_SCAL+1], t);
  t = block_sum<256>(bg, red);
  if(tid==0) atomicAdd(&ws[OFF_SCAL+2], t);
}

// pass 2, c-major: spatial-pool contexts
__global__ void k_ctx(const float* __restrict__ pS, const float* __restrict__ pT,
                      float* __restrict__ ws){
  __shared__ float red[16];
  const int n = blockIdx.x / CC, c = blockIdx.x % CC;
  const size_t base = ((size_t)n*CC + c)*HWP;
  const float* as = ws + OFF_A2 + n*HWP;
  const float* at = ws + OFF_A3 + n*HWP;
  float ss=0.f, st=0.f;
  for(int hw=threadIdx.x; hw<HWP; hw+=256){
    ss += pS[base+hw]*as[hw];
    st += pT[base+hw]*at[hw];
  }
  ss = block_sum<256>(ss, red);
  st = block_sum<256>(st, red);
  if(threadIdx.x==0){
    ws[OFF_CTXS + n*CC + c] = ss;
    ws[OFF_CTXT + n*CC + c] = st;
  }
}

// ---------------- WMMA MLP (f32 native) + final scalar combine ----------------
typedef float v2f __attribute__((ext_vector_type(2)));
typedef float v8f __attribute__((ext_vector_type(8)));

__global__ void __launch_bounds__(512)
k_mlp(const float* __restrict__ w1s, const float* __restrict__ b1s,
      const float* __restrict__ gs,  const float* __restrict__ bes,
      const float* __restrict__ w2s, const float* __restrict__ b2s,
      const float* __restrict__ w1t, const float* __restrict__ b1t,
      const float* __restrict__ gt,  const float* __restrict__ bet,
      const float* __restrict__ w2t, const float* __restrict__ b2t,
      float* __restrict__ ws, float* __restrict__ out){
  __shared__ float CT[2*16*CC];   // ctx (padded to 16 rows); reused as OUT after GEMM1
  __shared__ float HB[2*16*CH];   // hidden
  __shared__ float red[32];
  const int tid = threadIdx.x;

  for(int i=tid;i<2*16*CC;i+=512){
    int mat = i/(16*CC); int rem = i%(16*CC); int n = rem/CC; int c = rem%CC;
    CT[i] = (n<NB) ? ws[(mat ? OFF_CTXT : OFF_CTXS) + n*CC + c] : 0.f;
  }
  __syncthreads();

  const int lane = tid & 31, wave = tid >> 5;
  const int col = lane & 15, half = lane >> 4;   // D/A/B lane mapping per ISA tables

  // GEMM1: H = CTX(16x256) * W1^T(256x128) + b1  (16 waves: 2 mats x 8 j-tiles)
  {
    const int mat = wave >> 3, jt = wave & 7, jbase = jt*16;
    const float* w1 = mat ? w1t : w1s;
    const float* b1 = mat ? b1t : b1s;
    v8f acc = {0,0,0,0,0,0,0,0};
    for(int k=0;k<CC;k+=4){
      const int kk = k + 2*half;
      v2f a, b;
      a[0] = CT[mat*4096 + col*CC + kk];       // A[m=col][kk], A[m][kk+1]
      a[1] = CT[mat*4096 + col*CC + kk+1];
      const int j = jbase + col;
      b[0] = w1[j*CC + kk];                    // B[kk][n=col] = w1[j][kk]
      b[1] = w1[j*CC + kk+1];
      acc = __builtin_amdgcn_wmma_f32_16x16x4_f32(false, a, false, b,
                                                  (short)0, acc, false, false);
    }
    const float bias = b1[jbase+col];
#pragma unroll
    for(int i=0;i<8;++i){
      const int row = i + 8*half;              // D: VGPR i -> M=i / M=i+8
      HB[mat*2048 + row*CH + jbase + col] = acc[i] + bias;
    }
  }
  __syncthreads();

  // LayerNorm + ReLU over each of 32 rows (2 mats x 16 rows, 128 cols)
  {
    const int r = tid >> 4, sub = tid & 15;    // 16 lanes per row, 8 cols each
    const int mat = r >> 4, m = r & 15;
    float* h = &HB[mat*2048 + m*CH];
    float s = 0.f;
#pragma unroll
    for(int q=0;q<8;++q) s += h[sub*8+q];
#pragma unroll
    for(int o=8;o>0;o>>=1) s += __shfl_xor(s,o,16);
    const float mu = s * (1.0f/CH);
    float v = 0.f;
#pragma unroll
    for(int q=0;q<8;++q){ float d = h[sub*8+q]-mu; v += d*d; }
#pragma unroll
    for(int o=8;o>0;o>>=1) v += __shfl_xor(v,o,16);
    const float rs = rsqrtf(v*(1.0f/CH) + LNEPS);
    const float* g  = mat ? gt  : gs;
    const float* be = mat ? bet : bes;
#pragma unroll
    for(int q=0;q<8;++q){
      const int j = sub*8+q;
      h[j] = fmaxf((h[j]-mu)*rs*g[j] + be[j], 0.f);
    }
  }
  __syncthreads();

  // GEMM2: OUT = H'(16x128) * W2^T(128x256) + b2 (32 tiles, 2 per wave) -> CT region
  for(int t=wave; t<32; t+=16){
    const int mat = t >> 4, ctile = t & 15, cbase = ctile*16;
    const float* w2 = mat ? w2t : w2s;
    const float* b2 = mat ? b2t : b2s;
    v8f acc = {0,0,0,0,0,0,0,0};
    for(int k=0;k<CH;k+=4){
      const int kk = k + 2*half;
      v2f a, b;
      a[0] = HB[mat*2048 + col*CH + kk];
      a[1] = HB[mat*2048 + col*CH + kk+1];
      const int c = cbase + col;
      b[0] = w2[c*CH + kk];
      b[1] = w2[c*CH + kk+1];
      acc = __builtin_amdgcn_wmma_f32_16x16x4_f32(false, a, false, b,
                                                  (short)0, acc, false, false);
    }
    const float bias = b2[cbase+col];
#pragma unroll
    for(int i=0;i<8;++i){
      const int row = i + 8*half;
      CT[mat*4096 + row*CC + cbase + col] = acc[i] + bias;
    }
  }
  __syncthreads();

  // rela_loss = dsq + 2*sum(delta*dsum) + HW*sum(delta^2); final scalar combine
  float r1=0.f, r2=0.f;
  for(int i=tid;i<NB*CC;i+=512){
    const int n = i/CC, c = i%CC;
    const float delta = CT[n*CC + c] - CT[4096 + n*CC + c];
    r1 += delta * ws[OFF_DSUM + i];
    r2 += delta * delta;
  }
  r1 = block_sum<512>(r1, red);
  r2 = block_sum<512>(r2, red);
  if(tid==0){
    const float rela = ws[OFF_SCAL+0] + 2.0f*r1 + (float)HWP*r2;
    out[0] = (K_ALPHA*ws[OFF_SCAL+1] + K_BETA*ws[OFF_SCAL+2] +
              K_GAMMA*ws[OFF_SCAL+3] + K_LAMB*rela) * (1.0f/NB);
  }
}

// ---------------- launch ----------------
extern "C" void kernel_launch(void* const* d_in, const int* in_sizes, int n_in,
                              void* d_out, int out_size, void* d_ws, size_t ws_size,
                              hipStream_t stream) {
  (void)in_sizes; (void)n_in; (void)out_size; (void)ws_size;
  const float* pS    = (const float*)d_in[0];
  const float* pT    = (const float*)d_in[1];
  const float* boxes = (const float*)d_in[2];
  const float* wms   = (const float*)d_in[3];
  const float* bms   = (const float*)d_in[4];
  const float* wmt   = (const float*)d_in[5];
  const float* bmt   = (const float*)d_in[6];
  const float* w1s = (const float*)d_in[7];  const float* b1s = (const float*)d_in[8];
  const float* gs  = (const float*)d_in[9];  const float* bes = (const float*)d_in[10];
  const float* w2s = (const float*)d_in[11]; const float* b2s = (const float*)d_in[12];
  const float* w1t = (const float*)d_in[13]; const float* b1t = (const float*)d_in[14];
  const float* gt  = (const float*)d_in[15]; const float* bet = (const float*)d_in[16];
  const float* w2t = (const float*)d_in[17]; const float* b2t = (const float*)d_in[18];
  float* ws  = (float*)d_ws;
  float* out = (float*)d_out;

  hipLaunchKernelGGL(k_init,       dim3(1),        dim3(64),  0, stream, ws);
  hipLaunchKernelGGL(k_pass1_hw,   dim3(NB*TILES), dim3(256), 0, stream, pS,pT,wms,bms,wmt,bmt,ws);
  hipLaunchKernelGGL(k_pass1_nc,   dim3(NB*CC),    dim3(256), 0, stream, pS,pT,ws);
  hipLaunchKernelGGL(k_softmax_hw, dim3(NB*4),     dim3(256), 0, stream, ws);
  hipLaunchKernelGGL(k_softmax_c,  dim3(NB),       dim3(256), 0, stream, ws);
  hipLaunchKernelGGL(k_masks,      dim3(NB*TILES), dim3(256), 0, stream, boxes, ws);
  hipLaunchKernelGGL(k_fgbg,       dim3(NB*TILES), dim3(256), 0, stream, pS,pT,ws);
  hipLaunchKernelGGL(k_ctx,        dim3(NB*CC),    dim3(256), 0, stream, pS,pT,ws);
  hipLaunchKernelGGL(k_mlp,        dim3(1),        dim3(512), 0, stream,
                     w1s,b1s,gs,bes,w2s,b2s, w1t,b1t,gt,bet,w2t,b2t, ws, out);
}